// RMSLayerNorm_28836410425837
// MI455X (gfx1250) — compile-verified
//
#include <hip/hip_runtime.h>
#include <stdint.h>

typedef float v4f __attribute__((ext_vector_type(4)));

#define FEAT    1184        // 256*1 + 128*3 + 64*5 + 32*7
#define F4      296         // FEAT / 4 float4 chunks per node
#define NPB     8           // nodes per block: one wave32 per node
#define THREADS 256

__global__ __launch_bounds__(THREADS) void eqnorm_kernel(
    const float* __restrict__ x,
    const float* __restrict__ w0, const float* __restrict__ w1,
    const float* __restrict__ w2, const float* __restrict__ w3,
    const float* __restrict__ b0,
    float* __restrict__ out, int n_nodes)
{
    __shared__ __align__(16) float tbl_s[FEAT];       // fused per-feature weight w[m]
    __shared__ __align__(16) float tbl_b[256];        // bias: only l=0 features have one
    __shared__ __align__(16) float xbuf[NPB * FEAT];  // async-staged node rows

    const int tid  = threadIdx.x;
    const int wave = tid >> 5;
    const int lane = tid & 31;
    const int node = blockIdx.x * NPB + wave;
    const bool active = node < n_nodes;

    // ---- build fused scale table (segment boundaries: 256, 640, 960) ----
    for (int i = tid; i < FEAT; i += THREADS) {
        float s;
        if (i < 256)      { s = w0[i];             tbl_b[i] = b0[i]; }
        else if (i < 640) { s = w1[(i - 256) / 3]; }
        else if (i < 960) { s = w2[(i - 640) / 5]; }
        else              { s = w3[(i - 960) / 7]; }
        tbl_s[i] = s;
    }

    // ---- CDNA5 async global->LDS copy of this wave's node row ----
    // 9 x b128 (1152 floats) + 1 x b32 (32 floats) = 1184 floats, all lanes active.
    const uint32_t lds_row = (uint32_t)(uintptr_t)&xbuf[wave * FEAT];
    if (active) {
        const uint32_t row_byte = (uint32_t)node * (FEAT * 4u);
#pragma unroll
        for (int j = 0; j < 9; ++j) {
            uint32_t off  = (uint32_t)((j * 128 + lane * 4) * 4);
            uint32_t ldsa = lds_row + off;
            uint32_t voff = row_byte + off;
            asm volatile("global_load_async_to_lds_b128 %0, %1, %2 offset:0"
                         :: "v"(ldsa), "v"(voff), "s"(x) : "memory");
        }
        uint32_t offr  = (uint32_t)((1152 + lane) * 4);
        uint32_t ldsar = lds_row + offr;
        uint32_t voffr = row_byte + offr;
        asm volatile("global_load_async_to_lds_b32 %0, %1, %2 offset:0"
                     :: "v"(ldsar), "v"(voffr), "s"(x) : "memory");
    }

    __syncthreads();            // tables visible to all waves
    if (!active) return;

    // wait for THIS wave's async row (only this wave reads it -> no barrier needed)
    asm volatile("s_wait_asynccnt 0" ::: "memory");

    // ---- pass 1: pull row into registers, fused reductions ----
    // s1 = sum of l=0 scalars; s2 = sum coef*x^2 with coef = 1/(d^2 * 4)
    // centered-scalar correction done analytically: sum(x-mu)^2 = sum x^2 - 256 mu^2
    const v4f* __restrict__ xv = (const v4f*)&xbuf[wave * FEAT];
    v4f vals[10];
    float s1 = 0.0f, s2 = 0.0f;
#pragma unroll
    for (int kk = 0; kk < 10; ++kk) {
        int k = kk * 32 + lane;
        v4f v = {0.0f, 0.0f, 0.0f, 0.0f};
        if (kk < 9 || k < F4) v = xv[k];
        vals[kk] = v;
        // float4-chunk boundaries in k units: 64 (l=0), 160 (l=1), 240 (l=2);
        // only kk==7 straddles a boundary, the rest fold to constants.
        float coef = (k < 64)  ? 0.25f
                   : (k < 160) ? (1.0f / 36.0f)
                   : (k < 240) ? 0.01f
                   :             (1.0f / 196.0f);
        float sq = v.x * v.x + v.y * v.y + v.z * v.z + v.w * v.w;
        s2 = fmaf(coef, sq, s2);
        if (kk < 2) s1 += v.x + v.y + v.z + v.w;   // kk<2 <=> feature idx < 256
    }

    // wave32 butterfly reductions
#pragma unroll
    for (int m = 16; m > 0; m >>= 1) {
        s1 += __shfl_xor(s1, m, 32);
        s2 += __shfl_xor(s2, m, 32);
    }
    const float mu  = s1 * (1.0f / 256.0f);
    const float nrm = s2 - 64.0f * mu * mu;   // 0.25 * (sum x^2 - 256 mu^2) correction
    const float inv = rsqrtf(nrm + 1e-12f);

    // ---- pass 2: scale + bias, stream out non-temporally ----
    const v4f* __restrict__ sv = (const v4f*)tbl_s;
    const v4f* __restrict__ bv = (const v4f*)tbl_b;
    v4f* __restrict__ ov = (v4f*)(out + (size_t)node * FEAT);
#pragma unroll
    for (int kk = 0; kk < 10; ++kk) {
        int k = kk * 32 + lane;
        if (kk == 9 && k >= F4) continue;
        v4f v = vals[kk];
        v4f s = sv[k];
        v4f o;
        if (kk < 2) {                       // l=0: center, scale, add bias
            v4f b = bv[k];
            o = (v - mu) * (s * inv) + b;
        } else {                            // l>0: scale only
            o = v * (s * inv);
        }
        __builtin_nontemporal_store(o, ov + k);
    }
}

extern "C" void kernel_launch(void* const* d_in, const int* in_sizes, int n_in,
                              void* d_out, int out_size, void* d_ws, size_t ws_size,
                              hipStream_t stream) {
    (void)n_in; (void)out_size; (void)d_ws; (void)ws_size;
    const float* x  = (const float*)d_in[0];
    const float* w0 = (const float*)d_in[1];
    const float* w1 = (const float*)d_in[2];
    const float* w2 = (const float*)d_in[3];
    const float* w3 = (const float*)d_in[4];
    const float* b0 = (const float*)d_in[5];
    float* out = (float*)d_out;
    const int n_nodes = in_sizes[0] / FEAT;
    const int blocks  = (n_nodes + NPB - 1) / NPB;
    eqnorm_kernel<<<blocks, THREADS, 0, stream>>>(x, w0, w1, w2, w3, b0, out, n_nodes);
}